// GRUCell_60919816126900
// MI455X (gfx1250) — compile-verified
//
#include <hip/hip_runtime.h>
#include <math.h>

typedef __attribute__((ext_vector_type(2))) float v2f;
typedef __attribute__((ext_vector_type(4))) float v4f;
typedef __attribute__((ext_vector_type(8))) float v8f;

#define BROWS 16384
#define KDIM  512
#define NDIM  512

#define MT 64           // block tile rows
#define NT 64           // block tile cols
#define KC 64           // K chunk staged in LDS
#define NCHUNK (KDIM / KC)
#define LDA 68          // padded A row stride (floats): banks 4*lane -> conflict-free
#define BPS 160         // B pair-row stride (floats); 160 % 64 == 32 -> upper half-wave
                        // (K+2 pair-row) lands on banks 32..63: conflict-free b64 reads

__device__ __forceinline__ float fast_sigmoid(float x) {
    return 1.0f / (1.0f + __expf(-x));
}
__device__ __forceinline__ float fast_tanh(float x) {
    x = fminf(fmaxf(x, -15.0f), 15.0f);
    float e = __expf(2.0f * x);
    return (e - 1.0f) / (e + 1.0f);
}

__global__ __launch_bounds__(256) void gru_fused_wmma(
    const float* __restrict__ xin,   // (16384, 1024); x = cols [0,512)
    const float* __restrict__ wz,    // (512, 512)
    const float* __restrict__ bz,    // (512,)
    const float* __restrict__ wh,    // (512, 512)
    const float* __restrict__ bh,    // (512,)
    float* __restrict__ out)         // [B*1024 h_t_forward | B*512 h_t_seq]
{
    __shared__ float As [2][MT * LDA];          // A tile, row-major, padded (ping-pong)
    __shared__ float Bzs[2][(KC / 2) * BPS];    // w_z tile, K-pair (v2f) layout (ping-pong)
    __shared__ float Bhs[2][(KC / 2) * BPS];    // w_h tile, K-pair (v2f) layout (ping-pong)

    const int tid  = threadIdx.x;
    const int lane = tid & 31;
    const int wave = tid >> 5;          // 0..7
    const int wm   = wave & 3;          // wave's M sub-tile (16 rows)
    const int wn   = wave >> 2;         // wave's N sub-tile (32 cols)

    const int n0 = blockIdx.x * NT;
    const int m0 = blockIdx.y * MT;

    const int lr = tid >> 4;            // loader row 0..15
    const int lc = (tid & 15) << 2;     // loader col 0,4,...,60

    const int half = lane >> 4;         // 0: K-pair {k,k+1}, 1: {k+2,k+3}
    const int l15  = lane & 15;

    v8f accz0 = {}, accz1 = {}, acch0 = {}, acch1 = {};

    // register prefetch staging: A (4x v4f), B pairs (2 passes x 2 rows x 2 matrices)
    v4f ra[4];
    v4f rz[4], rh[4];

    // ---- prefetch chunk 0 into registers ----
    #pragma unroll
    for (int rr = 0; rr < 4; ++rr)
        ra[rr] = *(const v4f*)(xin + (size_t)(m0 + lr + rr * 16) * 1024 + lc);
    #pragma unroll
    for (int p = 0; p < 2; ++p) {
        const int pr   = (tid >> 4) + p * 16;
        const int krow = pr * 2;
        rz[2 * p]     = *(const v4f*)(wz + (size_t)krow * NDIM + n0 + lc);
        rz[2 * p + 1] = *(const v4f*)(wz + (size_t)(krow + 1) * NDIM + n0 + lc);
        rh[2 * p]     = *(const v4f*)(wh + (size_t)krow * NDIM + n0 + lc);
        rh[2 * p + 1] = *(const v4f*)(wh + (size_t)(krow + 1) * NDIM + n0 + lc);
    }

    int buf = 0;
    for (int c = 0; c < NCHUNK; ++c) {
        // ---- commit prefetched registers to LDS ping-pong buffer buf ----
        #pragma unroll
        for (int rr = 0; rr < 4; ++rr)
            *(v4f*)(&As[buf][0] + (lr + rr * 16) * LDA + lc) = ra[rr];
        #pragma unroll
        for (int p = 0; p < 2; ++p) {
            const int pr = (tid >> 4) + p * 16;
            float* dz = &Bzs[buf][0] + pr * BPS + lc * 2;
            float* dh = &Bhs[buf][0] + pr * BPS + lc * 2;
            #pragma unroll
            for (int j = 0; j < 4; ++j) {
                v2f pz = { rz[2 * p][j], rz[2 * p + 1][j] };
                v2f ph = { rh[2 * p][j], rh[2 * p + 1][j] };
                *(v2f*)(dz + j * 2) = pz;
                *(v2f*)(dh + j * 2) = ph;
            }
        }
        __syncthreads();    // single barrier per chunk (ping-pong makes this sufficient)

        // ---- issue global loads for chunk c+1 (hidden under the WMMA stream) ----
        if (c + 1 < NCHUNK) {
            const int k0n = (c + 1) * KC;
            #pragma unroll
            for (int rr = 0; rr < 4; ++rr)
                ra[rr] = *(const v4f*)(xin + (size_t)(m0 + lr + rr * 16) * 1024 + k0n + lc);
            #pragma unroll
            for (int p = 0; p < 2; ++p) {
                const int pr   = (tid >> 4) + p * 16;
                const int krow = k0n + pr * 2;
                rz[2 * p]     = *(const v4f*)(wz + (size_t)krow * NDIM + n0 + lc);
                rz[2 * p + 1] = *(const v4f*)(wz + (size_t)(krow + 1) * NDIM + n0 + lc);
                rh[2 * p]     = *(const v4f*)(wh + (size_t)krow * NDIM + n0 + lc);
                rh[2 * p + 1] = *(const v4f*)(wh + (size_t)(krow + 1) * NDIM + n0 + lc);
            }
            if (c + 2 < NCHUNK)
                __builtin_prefetch(xin + (size_t)(m0 + lr) * 1024 + (c + 2) * KC + lc, 0, 0);
        }

        // ---- compute: V_WMMA_F32_16X16X4_F32 over LDS ping-pong buffer buf ----
        const float* Arow = &As[buf][0] + (wm * 16 + l15) * LDA;
        const float* Bzb  = &Bzs[buf][0];
        const float* Bhb  = &Bhs[buf][0];
        const int nbase = wn * 32 + l15;
        #pragma unroll
        for (int k = 0; k < KC; k += 4) {
            const int kk = k + half * 2;
            const int r  = kk >> 1;                       // pair-row for this half-wave
            v2f a   = *(const v2f*)(Arow + kk);           // one b64
            v2f vz0 = *(const v2f*)(Bzb + r * BPS + nbase * 2);
            v2f vz1 = *(const v2f*)(Bzb + r * BPS + (nbase + 16) * 2);
            v2f vh0 = *(const v2f*)(Bhb + r * BPS + nbase * 2);
            v2f vh1 = *(const v2f*)(Bhb + r * BPS + (nbase + 16) * 2);

            accz0 = __builtin_amdgcn_wmma_f32_16x16x4_f32(false, a, false, vz0, (short)0, accz0, false, false);
            accz1 = __builtin_amdgcn_wmma_f32_16x16x4_f32(false, a, false, vz1, (short)0, accz1, false, false);
            acch0 = __builtin_amdgcn_wmma_f32_16x16x4_f32(false, a, false, vh0, (short)0, acch0, false, false);
            acch1 = __builtin_amdgcn_wmma_f32_16x16x4_f32(false, a, false, vh1, (short)0, acch1, false, false);
        }
        buf ^= 1;
    }

    // ---- epilogue: bias + sigmoid(z)*tanh(h), write all three output regions ----
    const int colA = n0 + wn * 32 + l15;        // N sub-tile 0 column for this lane
    const int colB = colA + 16;                 // N sub-tile 1 column
    const float bz0 = bz[colA], bz1 = bz[colB];
    const float bh0 = bh[colA], bh1 = bh[colB];

    const size_t seq_base = (size_t)BROWS * 1024;

    #pragma unroll
    for (int v = 0; v < 8; ++v) {
        const int m_loc = v + half * 8;         // C/D layout: VGPR v, lanes>=16 -> M = v+8
        const int row   = m0 + wm * 16 + m_loc;
        const size_t ro = (size_t)row * 1024;
        const size_t so = seq_base + (size_t)row * 512;

        float s0 = fast_sigmoid(accz0[v] + bz0) * fast_tanh(acch0[v] + bh0);
        float s1 = fast_sigmoid(accz1[v] + bz1) * fast_tanh(acch1[v] + bh1);

        out[ro + colA]       = s0;              // h_t_forward[:, 0:512]
        out[ro + 512 + colA] = 0.0f;            // h_t_forward[:, 512:1024] == out_mask == 0
        out[so + colA]       = s0;              // h_t_seq
        out[ro + colB]       = s1;
        out[ro + 512 + colB] = 0.0f;
        out[so + colB]       = s1;
    }
}

extern "C" void kernel_launch(void* const* d_in, const int* in_sizes, int n_in,
                              void* d_out, int out_size, void* d_ws, size_t ws_size,
                              hipStream_t stream) {
    // setup_inputs order: inputs(0) h(1) w_z(2) u_z(3) b_z(4) w_r(5) u_r(6) b_r(7) w_h(8) u_h(9) b_h(10)
    const float* xin = (const float*)d_in[0];
    const float* wz  = (const float*)d_in[2];
    const float* bz  = (const float*)d_in[4];
    const float* wh  = (const float*)d_in[8];
    const float* bh  = (const float*)d_in[10];
    float* out = (float*)d_out;

    dim3 grid(NDIM / NT, BROWS / MT);   // (8, 256)
    dim3 block(256);
    gru_fused_wmma<<<grid, block, 0, stream>>>(xin, wz, bz, wh, bh, out);
}